// KMeans_53523882443508
// MI455X (gfx1250) — compile-verified
//
#include <hip/hip_runtime.h>
#include <math.h>

// ---------------------------------------------------------------------------
// KMeans (Lloyd) + per-cluster top-15 softmax feature aggregation, CDNA5.
// Distance argmax done with V_WMMA_F32_16X16X4_F32 via the augmentation trick:
//   A row  (point)    = [px, py, pz, 1]
//   B col  (centroid) = [2cx, 2cy, 2cz, -|c|^2]
//   D = 2 p.c - |c|^2   (same row-argmax as full euc_sim)
// ---------------------------------------------------------------------------

typedef __attribute__((ext_vector_type(2))) float v2f;
typedef __attribute__((ext_vector_type(8))) float v8f;

#define BATCH   2
#define NPTS    65536
#define NCLUST  512
#define NFEAT   128
#define KNEIGH  15
#define MAXITER 15
#define TOLSQ   1e-4f
#define NEGINF  (-1e30f)
#define CAND    4096

// ---------------- init: working centroids, augmented centroids, zero state --
__global__ void km_init(const float* __restrict__ cen_in,
                        float* __restrict__ c, float* __restrict__ caug,
                        float* __restrict__ sums, int* __restrict__ cnt,
                        int* __restrict__ flags) {
  int b = blockIdx.x, m = threadIdx.x;
  const float* s = cen_in + ((size_t)(b * NCLUST + m) * 3);
  float x = s[0], y = s[1], z = s[2];
  float* cd = c + ((size_t)(b * NCLUST + m) * 3);
  cd[0] = x; cd[1] = y; cd[2] = z;
  float* ca = caug + ((size_t)(b * NCLUST + m) * 4);
  ca[0] = 2.f * x; ca[1] = 2.f * y; ca[2] = 2.f * z;
  ca[3] = -(x * x + y * y + z * z);
  float* sm = sums + ((size_t)(b * NCLUST + m) * 3);
  sm[0] = 0.f; sm[1] = 0.f; sm[2] = 0.f;
  cnt[b * NCLUST + m] = 0;
  if (m == 0) flags[b] = 0;
}

// ---------------- assignment + scatter-mean accumulation (WMMA) -------------
// 256 threads = 8 waves; each wave owns a 16-point tile vs all 512 clusters.
__global__ void km_assign(const float* __restrict__ points,
                          const float* __restrict__ caug,
                          int* __restrict__ labels,
                          float* __restrict__ sums, int* __restrict__ cnt,
                          const int* __restrict__ flags) {
  int blk   = blockIdx.x;          // 0..1023  (512 blocks per batch)
  int b     = blk >> 9;
  int pbase = (blk & 511) * 128;   // 128 points per block
  if (flags[b]) return;            // converged: uniform no-op

  __shared__ float s_caug[NCLUST * 4];           // 8 KB augmented centroids
  __shared__ float s_sx[NCLUST], s_sy[NCLUST], s_sz[NCLUST];
  __shared__ int   s_cn[NCLUST];
  __shared__ int   s_lab[128];

  int tid = threadIdx.x;

  // Stage centroids into LDS (2 x float4 per thread) and zero accumulators.
  const float4* gca = (const float4*)(caug + (size_t)b * NCLUST * 4);
  float4* sca = (float4*)s_caug;
  sca[tid]       = gca[tid];
  sca[tid + 256] = gca[tid + 256];
  s_sx[tid] = 0.f; s_sy[tid] = 0.f; s_sz[tid] = 0.f; s_cn[tid] = 0;
  s_sx[tid + 256] = 0.f; s_sy[tid + 256] = 0.f; s_sz[tid + 256] = 0.f;
  s_cn[tid + 256] = 0;
  __syncthreads();

  int lane  = tid & 31;
  int wave  = tid >> 5;
  int tbase = pbase + wave * 16;

  // A (16x4 f32): lanes 0-15 hold K=0,1 (px,py); lanes 16-31 hold K=2,3 (pz,1)
  int row = lane & 15;
  const float* P = points + ((size_t)b * NPTS + (size_t)(tbase + row)) * 3;
  v2f a;
  if (lane < 16) { a.x = P[0]; a.y = P[1]; }
  else           { a.x = P[2]; a.y = 1.0f; }

  float bs[8];
  int   bi[8];
  #pragma unroll
  for (int r = 0; r < 8; ++r) { bs[r] = -3.0e38f; bi[r] = 0; }

  int nlow = lane & 15;
  int off  = (lane < 16) ? 0 : 2;   // low lanes K=0,1 ; high lanes K=2,3

  #pragma unroll
  for (int t = 0; t < 32; ++t) {
    int n = (t << 4) + nlow;        // this lane's cluster column
    float2 bb = *(const float2*)(&s_caug[(n << 2) + off]);  // ds_load_b64
    v2f bm; bm.x = bb.x; bm.y = bb.y;
    v8f acc = {};
    acc = __builtin_amdgcn_wmma_f32_16x16x4_f32(false, a, false, bm,
                                                (short)0, acc, false, false);
    // D slot (VGPR r, lane l) = row (r or r+8), col l&15 -> cluster n.
    #pragma unroll
    for (int r = 0; r < 8; ++r) {
      float sv = acc[r];
      if (sv > bs[r]) { bs[r] = sv; bi[r] = n; }  // strict >: first-max wins
    }
  }

  // Row argmax: reduce across the 16 lanes of each half-wave.
  #pragma unroll
  for (int r = 0; r < 8; ++r) {
    float sv = bs[r]; int iv = bi[r];
    #pragma unroll
    for (int d = 1; d < 16; d <<= 1) {
      float s2 = __shfl_xor(sv, d, 32);
      int   i2 = __shfl_xor(iv, d, 32);
      if (s2 > sv || (s2 == sv && i2 < iv)) { sv = s2; iv = i2; }
    }
    if ((lane & 15) == 0) {
      int m_row = r + ((lane >> 4) << 3);  // lane0 -> row r ; lane16 -> row r+8
      s_lab[wave * 16 + m_row] = iv;
    }
  }
  __syncthreads();

  // Per-point scatter into LDS accumulators + write labels.
  if (tid < 128) {
    int pidx = pbase + tid;
    int lab  = s_lab[tid];
    const float* Q = points + ((size_t)b * NPTS + pidx) * 3;
    atomicAdd(&s_sx[lab], Q[0]);
    atomicAdd(&s_sy[lab], Q[1]);
    atomicAdd(&s_sz[lab], Q[2]);
    atomicAdd(&s_cn[lab], 1);
    labels[(size_t)b * NPTS + pidx] = lab;
  }
  __syncthreads();

  // Flush block-local sums to global.
  for (int m = tid; m < NCLUST; m += 256) {
    int cn = s_cn[m];
    if (cn) {
      float* gs = sums + ((size_t)b * NCLUST + m) * 3;
      atomicAdd(&gs[0], s_sx[m]);
      atomicAdd(&gs[1], s_sy[m]);
      atomicAdd(&gs[2], s_sz[m]);
      atomicAdd(&cnt[b * NCLUST + m], cn);
    }
  }
}

// ---------------- centroid update + convergence flag ------------------------
__global__ void km_update(float* __restrict__ c, float* __restrict__ caug,
                          float* __restrict__ sums, int* __restrict__ cnt,
                          int* __restrict__ flags) {
  int b = blockIdx.x, m = threadIdx.x;
  if (flags[b]) return;
  __shared__ float red[NCLUST];
  size_t o3 = ((size_t)b * NCLUST + m) * 3;
  float n_  = (float)cnt[b * NCLUST + m];
  float inv = 1.0f / (n_ + 1e-8f);
  float nx = sums[o3 + 0] * inv, ny = sums[o3 + 1] * inv, nz = sums[o3 + 2] * inv;
  float dx = nx - c[o3 + 0], dy = ny - c[o3 + 1], dz = nz - c[o3 + 2];
  c[o3 + 0] = nx; c[o3 + 1] = ny; c[o3 + 2] = nz;
  float* ca = caug + ((size_t)b * NCLUST + m) * 4;
  ca[0] = 2.f * nx; ca[1] = 2.f * ny; ca[2] = 2.f * nz;
  ca[3] = -(nx * nx + ny * ny + nz * nz);
  sums[o3 + 0] = 0.f; sums[o3 + 1] = 0.f; sums[o3 + 2] = 0.f;
  cnt[b * NCLUST + m] = 0;
  red[m] = dx * dx + dy * dy + dz * dz;
  __syncthreads();
  for (int s = 256; s > 0; s >>= 1) {
    if (m < s) red[m] += red[m + s];
    __syncthreads();
  }
  if (m == 0 && red[0] <= TOLSQ) flags[b] = 1;
}

// ---------------- output copies --------------------------------------------
__global__ void km_copy_cc(const float* __restrict__ c, float* __restrict__ out) {
  int i = blockIdx.x * blockDim.x + threadIdx.x;
  if (i < BATCH * NCLUST * 3) out[i] = c[i];
}
__global__ void km_copy_lbl(const int* __restrict__ labels, float* __restrict__ out) {
  int i = blockIdx.x * blockDim.x + threadIdx.x;
  if (i < BATCH * NPTS) out[i] = (float)labels[i];
}

// ---------------- per-cluster top-15 + softmax + feature gather -------------
// One block per (batch, cluster). Scores: 2 p.c - |p|^2  (softmax is invariant
// to the per-cluster constant -|c|^2, and ranking too).
__global__ void km_topk_cf(const float* __restrict__ points,
                           const float* __restrict__ features,
                           const float* __restrict__ c,
                           const int* __restrict__ labels,
                           float* __restrict__ out_cf) {
  int b   = blockIdx.x >> 9;
  int m   = blockIdx.x & (NCLUST - 1);
  int tid = threadIdx.x;

  __shared__ float sc[CAND];
  __shared__ int   si[CAND];
  __shared__ int   scnt;
  __shared__ float rS[256];
  __shared__ int   rI[256];
  __shared__ int   rP[256];
  __shared__ float w[KNEIGH];
  __shared__ int   sel[KNEIGH];

  const float* cm = c + ((size_t)b * NCLUST + m) * 3;
  float cx = cm[0], cy = cm[1], cz = cm[2];
  if (tid == 0) scnt = 0;
  __syncthreads();

  const int*   lab = labels + (size_t)b * NPTS;
  const float* P   = points + (size_t)b * NPTS * 3;
  for (int i = tid; i < NPTS; i += 256) {
    if (lab[i] == m) {
      float x = P[i * 3 + 0], y = P[i * 3 + 1], z = P[i * 3 + 2];
      float s = 2.f * (x * cx + y * cy + z * cz) - (x * x + y * y + z * z);
      int slot = atomicAdd(&scnt, 1);
      if (slot < CAND) { sc[slot] = s; si[slot] = i; }
    }
  }
  __syncthreads();
  int cnt = scnt; if (cnt > CAND) cnt = CAND;

  // 15 rounds of parallel argmax extraction (tie -> lower point index).
  for (int k = 0; k < KNEIGH; ++k) {
    float bsV = -3.0e38f; int bsI = 0x7fffffff; int bsP = -1;
    for (int j = tid; j < cnt; j += 256) {
      float s = sc[j]; int pi = si[j];
      if (s > bsV || (s == bsV && pi < bsI)) { bsV = s; bsI = pi; bsP = j; }
    }
    rS[tid] = bsV; rI[tid] = bsI; rP[tid] = bsP;
    __syncthreads();
    for (int st = 128; st > 0; st >>= 1) {
      if (tid < st) {
        float s2 = rS[tid + st]; int i2 = rI[tid + st];
        if (s2 > rS[tid] || (s2 == rS[tid] && i2 < rI[tid])) {
          rS[tid] = s2; rI[tid] = i2; rP[tid] = rP[tid + st];
        }
      }
      __syncthreads();
    }
    if (tid == 0) {
      if (k < cnt) { sel[k] = rI[0]; w[k] = rS[0]; sc[rP[0]] = -3.0e38f; }
      else         { sel[k] = k;     w[k] = NEGINF; }  // zero weight (or uniform if empty)
    }
    __syncthreads();
  }

  if (tid == 0) {
    float mx = w[0];                 // first extraction is the max
    float sum = 0.f;
    for (int k = 0; k < KNEIGH; ++k) { float e = expf(w[k] - mx); w[k] = e; sum += e; }
    float invs = 1.f / sum;
    for (int k = 0; k < KNEIGH; ++k) w[k] *= invs;
  }
  __syncthreads();

  if (tid < NFEAT) {
    const float* fr = features + ((size_t)b * NFEAT + tid) * NPTS;
    float acc = 0.f;
    #pragma unroll
    for (int k = 0; k < KNEIGH; ++k) acc += w[k] * fr[sel[k]];
    out_cf[((size_t)b * NFEAT + tid) * NCLUST + m] = acc;
  }
}

// ---------------------------------------------------------------------------
extern "C" void kernel_launch(void* const* d_in, const int* in_sizes, int n_in,
                              void* d_out, int out_size, void* d_ws, size_t ws_size,
                              hipStream_t stream) {
  const float* points    = (const float*)d_in[0];  // [B, N, 3]
  const float* features  = (const float*)d_in[1];  // [B, F, N]
  const float* centroids = (const float*)d_in[2];  // [B, M, 3]
  float* out = (float*)d_out;  // cc[B,M,3] | cf[B,F,M] | lbl[B,N] (as float)

  // Workspace carve-up (all 16B-aligned segments).
  float* c      = (float*)d_ws;                 // B*M*3
  float* caug   = c + BATCH * NCLUST * 3;       // B*M*4
  float* sums   = caug + BATCH * NCLUST * 4;    // B*M*3
  int*   cnt    = (int*)(sums + BATCH * NCLUST * 3);  // B*M
  int*   labels = cnt + BATCH * NCLUST;         // B*N
  int*   flags  = labels + BATCH * NPTS;        // B

  km_init<<<BATCH, NCLUST, 0, stream>>>(centroids, c, caug, sums, cnt, flags);

  for (int it = 0; it < MAXITER; ++it) {
    km_assign<<<dim3(BATCH * NPTS / 128), 256, 0, stream>>>(
        points, caug, labels, sums, cnt, flags);
    km_update<<<BATCH, NCLUST, 0, stream>>>(c, caug, sums, cnt, flags);
  }

  km_copy_cc<<<(BATCH * NCLUST * 3 + 255) / 256, 256, 0, stream>>>(c, out);
  km_topk_cf<<<BATCH * NCLUST, 256, 0, stream>>>(
      points, features, c, labels, out + BATCH * NCLUST * 3);
  km_copy_lbl<<<(BATCH * NPTS + 255) / 256, 256, 0, stream>>>(
      labels, out + BATCH * NCLUST * 3 + BATCH * NFEAT * NCLUST);
}